// SPGANGenerator_55130200211663
// MI455X (gfx1250) — compile-verified
//
#include <hip/hip_runtime.h>
#include <math.h>

typedef float v2f __attribute__((ext_vector_type(2)));
typedef float v8f __attribute__((ext_vector_type(8)));

#define NEG_A 0.01f
#define NEG_B 0.2f
#define KNN 10
#define NPTS 2048
#define NBATCH 8

// ---------------------------------------------------------------------------
// Generic fused GEMM on the fp32 WMMA pipe:  out[z,m,n] = act((W @ Bop) + bias)
// MODE 0: B element = Bp[r*sK1 + col]                    (plain channel-major)
// MODE 1: B element = Bp[(r/K2)*sK1 + col*sN + (r%K2)]   (split reduction; K2>4)
// MODE 2: edge gather: re=r+co; re<C ? X[re*N+nn] : X[(re-C)*N+nbr]-X[(re-C)*N+nn]
// Each wave computes a 16(M)x64(N) tile (four accumulators sharing one A
// fragment) with V_WMMA_F32_16X16X4_F32 chained over K in steps of 4.
// REQUIREMENTS (guaranteed by the host): Nc % 512 == 0, M % 16 == 0.
// Main K loop is branchless (unconditional loads); K%4 tail uses clamped
// indices + selects (no divergent branches around the WMMA).
// ---------------------------------------------------------------------------
template <int MODE>
__global__ __launch_bounds__(256) void gemm_wmma_kernel(
    const float* __restrict__ A, int Astride, int Ktot, int M,
    const float* __restrict__ Bb, long long bBatch, int sK1, int sN, int K2,
    const float* __restrict__ X, int Cedge, int co, int Npts, int Knn,
    const int* __restrict__ idxp,
    float* __restrict__ out, long long oBatch, int Nc,
    const float* __restrict__ bias, int biasBatch,
    const float* __restrict__ gamma, const float* __restrict__ beta,
    float bnr, float slope)
{
    const int lane = threadIdx.x & 31;
    const int wave = threadIdx.x >> 5;
    const int z    = blockIdx.z;
    const int m0   = blockIdx.y * 16;
    const int col0 = blockIdx.x * 512 + wave * 64 + (lane & 15);
    const int col1 = col0 + 16, col2 = col0 + 32, col3 = col0 + 48;
    const int kb   = (lane < 16) ? 0 : 2;   // A/B frag: lanes 0-15 -> K+0/K+1, 16-31 -> K+2/K+3
    const int rowA = lane & 15;

    const float* Arow = A + (long long)(m0 + rowA) * Astride + kb;
    const float* Bp   = Bb ? (Bb + (long long)z * bBatch) : nullptr;

    // Per-column precomputation
    const float* bc0 = nullptr; const float* bc1 = nullptr;        // MODE 0/1
    const float* bc2 = nullptr; const float* bc3 = nullptr;
    const float* ctr0 = nullptr; const float* nb0 = nullptr;       // MODE 2
    const float* ctr1 = nullptr; const float* nb1 = nullptr;
    const float* ctr2 = nullptr; const float* nb2 = nullptr;
    const float* ctr3 = nullptr; const float* nb3 = nullptr;
    if (MODE == 0) {
        const long long kboff = (long long)kb * sK1;
        bc0 = Bp + col0 + kboff;  bc1 = Bp + col1 + kboff;
        bc2 = Bp + col2 + kboff;  bc3 = Bp + col3 + kboff;
    } else if (MODE == 1) {
        bc0 = Bp + (long long)col0 * sN;  bc1 = Bp + (long long)col1 * sN;
        bc2 = Bp + (long long)col2 * sN;  bc3 = Bp + (long long)col3 * sN;
    } else {
        int nn0 = col0 / Knn, nn1 = col1 / Knn, nn2 = col2 / Knn, nn3 = col3 / Knn;
        ctr0 = X + nn0;  nb0 = X + idxp[col0];
        ctr1 = X + nn1;  nb1 = X + idxp[col1];
        ctr2 = X + nn2;  nb2 = X + idxp[col2];
        ctr3 = X + nn3;  nb3 = X + idxp[col3];
    }

    v8f acc0 = {0.f, 0.f, 0.f, 0.f, 0.f, 0.f, 0.f, 0.f};
    v8f acc1 = {0.f, 0.f, 0.f, 0.f, 0.f, 0.f, 0.f, 0.f};
    v8f acc2 = {0.f, 0.f, 0.f, 0.f, 0.f, 0.f, 0.f, 0.f};
    v8f acc3 = {0.f, 0.f, 0.f, 0.f, 0.f, 0.f, 0.f, 0.f};

    const int kMain = Ktot & ~3;

    // MODE 1 incremental (k1,k2) for r0 = k + kb and r1 = r0 + 1 (assumes K2 > 4)
    int k1a = 0, k2a = kb;

    for (int k = 0; k < kMain; k += 4) {
        v2f a, bf0, bf1, bf2, bf3;
        a.x = Arow[k];
        a.y = Arow[k + 1];
        if (MODE == 0) {
            const long long o0 = (long long)k * sK1;
            bf0.x = bc0[o0];  bf0.y = bc0[o0 + sK1];
            bf1.x = bc1[o0];  bf1.y = bc1[o0 + sK1];
            bf2.x = bc2[o0];  bf2.y = bc2[o0 + sK1];
            bf3.x = bc3[o0];  bf3.y = bc3[o0 + sK1];
        } else if (MODE == 1) {
            const long long e0 = (long long)k1a * sK1 + k2a;
            const long long e1 = (k2a == K2 - 1) ? (long long)(k1a + 1) * sK1
                                                 : e0 + 1;
            bf0.x = bc0[e0];  bf0.y = bc0[e1];
            bf1.x = bc1[e0];  bf1.y = bc1[e1];
            bf2.x = bc2[e0];  bf2.y = bc2[e1];
            bf3.x = bc3[e0];  bf3.y = bc3[e1];
            k2a += 4;
            if (k2a >= K2) { k2a -= K2; ++k1a; }
        } else {
            const int re0 = k + kb + co, re1 = re0 + 1;
            const long long g0 = (long long)re0 * Npts;
            const long long g1 = (long long)re1 * Npts;
            const long long d0 = (long long)(re0 - Cedge) * Npts;
            const long long d1 = (long long)(re1 - Cedge) * Npts;
            bf0.x = (re0 < Cedge) ? ctr0[g0] : (nb0[d0] - ctr0[d0]);
            bf0.y = (re1 < Cedge) ? ctr0[g1] : (nb0[d1] - ctr0[d1]);
            bf1.x = (re0 < Cedge) ? ctr1[g0] : (nb1[d0] - ctr1[d0]);
            bf1.y = (re1 < Cedge) ? ctr1[g1] : (nb1[d1] - ctr1[d1]);
            bf2.x = (re0 < Cedge) ? ctr2[g0] : (nb2[d0] - ctr2[d0]);
            bf2.y = (re1 < Cedge) ? ctr2[g1] : (nb2[d1] - ctr2[d1]);
            bf3.x = (re0 < Cedge) ? ctr3[g0] : (nb3[d0] - ctr3[d0]);
            bf3.y = (re1 < Cedge) ? ctr3[g1] : (nb3[d1] - ctr3[d1]);
        }
        acc0 = __builtin_amdgcn_wmma_f32_16x16x4_f32(false, a, false, bf0, (short)0, acc0, false, false);
        acc1 = __builtin_amdgcn_wmma_f32_16x16x4_f32(false, a, false, bf1, (short)0, acc1, false, false);
        acc2 = __builtin_amdgcn_wmma_f32_16x16x4_f32(false, a, false, bf2, (short)0, acc2, false, false);
        acc3 = __builtin_amdgcn_wmma_f32_16x16x4_f32(false, a, false, bf3, (short)0, acc3, false, false);
    }

    if (kMain < Ktot) {   // masked tail (clamped index + select; no branches)
        const int r0 = kMain + kb, r1 = r0 + 1;
        const int s0 = (r0 < Ktot) ? r0 : (Ktot - 1);
        const int s1 = (r1 < Ktot) ? r1 : (Ktot - 1);
        const bool v0 = (r0 < Ktot), v1 = (r1 < Ktot);
        const float* Ar = A + (long long)(m0 + rowA) * Astride;
        v2f a, bf0, bf1, bf2, bf3;
        a.x = v0 ? Ar[s0] : 0.f;
        a.y = v1 ? Ar[s1] : 0.f;
        if (MODE == 0) {
            const float* c0 = Bp + col0; const float* c1 = Bp + col1;
            const float* c2 = Bp + col2; const float* c3 = Bp + col3;
            const long long o0 = (long long)s0 * sK1, o1 = (long long)s1 * sK1;
            bf0.x = v0 ? c0[o0] : 0.f;  bf0.y = v1 ? c0[o1] : 0.f;
            bf1.x = v0 ? c1[o0] : 0.f;  bf1.y = v1 ? c1[o1] : 0.f;
            bf2.x = v0 ? c2[o0] : 0.f;  bf2.y = v1 ? c2[o1] : 0.f;
            bf3.x = v0 ? c3[o0] : 0.f;  bf3.y = v1 ? c3[o1] : 0.f;
        } else if (MODE == 1) {
            const int q0 = s0 / K2, q1 = s1 / K2;
            const long long e0 = (long long)q0 * sK1 + (s0 - q0 * K2);
            const long long e1 = (long long)q1 * sK1 + (s1 - q1 * K2);
            bf0.x = v0 ? bc0[e0] : 0.f;  bf0.y = v1 ? bc0[e1] : 0.f;
            bf1.x = v0 ? bc1[e0] : 0.f;  bf1.y = v1 ? bc1[e1] : 0.f;
            bf2.x = v0 ? bc2[e0] : 0.f;  bf2.y = v1 ? bc2[e1] : 0.f;
            bf3.x = v0 ? bc3[e0] : 0.f;  bf3.y = v1 ? bc3[e1] : 0.f;
        } else {
            const int re0 = s0 + co, re1 = s1 + co;
            const long long g0 = (long long)re0 * Npts, g1 = (long long)re1 * Npts;
            const long long d0 = (long long)(re0 - Cedge) * Npts;
            const long long d1 = (long long)(re1 - Cedge) * Npts;
            float t00 = (re0 < Cedge) ? ctr0[g0] : (nb0[d0] - ctr0[d0]);
            float t01 = (re1 < Cedge) ? ctr0[g1] : (nb0[d1] - ctr0[d1]);
            float t10 = (re0 < Cedge) ? ctr1[g0] : (nb1[d0] - ctr1[d0]);
            float t11 = (re1 < Cedge) ? ctr1[g1] : (nb1[d1] - ctr1[d1]);
            float t20 = (re0 < Cedge) ? ctr2[g0] : (nb2[d0] - ctr2[d0]);
            float t21 = (re1 < Cedge) ? ctr2[g1] : (nb2[d1] - ctr2[d1]);
            float t30 = (re0 < Cedge) ? ctr3[g0] : (nb3[d0] - ctr3[d0]);
            float t31 = (re1 < Cedge) ? ctr3[g1] : (nb3[d1] - ctr3[d1]);
            bf0.x = v0 ? t00 : 0.f;  bf0.y = v1 ? t01 : 0.f;
            bf1.x = v0 ? t10 : 0.f;  bf1.y = v1 ? t11 : 0.f;
            bf2.x = v0 ? t20 : 0.f;  bf2.y = v1 ? t21 : 0.f;
            bf3.x = v0 ? t30 : 0.f;  bf3.y = v1 ? t31 : 0.f;
        }
        acc0 = __builtin_amdgcn_wmma_f32_16x16x4_f32(false, a, false, bf0, (short)0, acc0, false, false);
        acc1 = __builtin_amdgcn_wmma_f32_16x16x4_f32(false, a, false, bf1, (short)0, acc1, false, false);
        acc2 = __builtin_amdgcn_wmma_f32_16x16x4_f32(false, a, false, bf2, (short)0, acc2, false, false);
        acc3 = __builtin_amdgcn_wmma_f32_16x16x4_f32(false, a, false, bf3, (short)0, acc3, false, false);
    }

    // Epilogue. C/D layout: vgpr v -> row v (lanes 0-15) / row v+8 (lanes 16-31)
    const int mHalf = (lane < 16) ? 0 : 8;
    float* orow = out + (long long)z * oBatch;
#pragma unroll
    for (int v = 0; v < 8; ++v) {
        const int m = m0 + mHalf + v;
        float y0 = acc0[v], y1 = acc1[v], y2 = acc2[v], y3 = acc3[v];
        if (bias)  { float bv = bias[z * biasBatch + m]; y0 += bv; y1 += bv; y2 += bv; y3 += bv; }
        if (gamma) { float g = gamma[m] * bnr, be = beta[m];
                     y0 = y0 * g + be; y1 = y1 * g + be; y2 = y2 * g + be; y3 = y3 * g + be; }
        y0 = (y0 >= 0.f) ? y0 : slope * y0;
        y1 = (y1 >= 0.f) ? y1 : slope * y1;
        y2 = (y2 >= 0.f) ? y2 : slope * y2;
        y3 = (y3 >= 0.f) ? y3 : slope * y3;
        float* p = orow + (long long)m * Nc;
        p[col0] = y0; p[col1] = y1; p[col2] = y2; p[col3] = y3;
    }
}

// ---------------------------------------------------------------------------
// Fused KNN: one wave per query point. Per-lane register top-11 lists (strict
// lexicographic (d, idx) to mimic stable argsort), merged with 11 rounds of
// wave-wide shfl_xor min; round 0 (self) dropped -> outputs argsort[...,1:11].
// ---------------------------------------------------------------------------
template <int C>
__global__ __launch_bounds__(32) void knn_kernel(
    const float* __restrict__ X,   // (B, C, N) channel-major
    const float* __restrict__ sq,  // (B, N)
    int* __restrict__ idxOut,      // (B, N, 10)
    int N)
{
    const int n = blockIdx.x, b = blockIdx.y, lane = threadIdx.x;
    const float* Xb  = X  + (long long)b * C * N;
    const float* sqb = sq + (long long)b * N;

    float q[C];
#pragma unroll
    for (int c = 0; c < C; ++c) q[c] = Xb[c * N + n];
    const float qs = sqb[n];

    const int TOP = 11;
    float bd[TOP]; int bi[TOP];
#pragma unroll
    for (int t = 0; t < TOP; ++t) { bd[t] = 3.4e38f; bi[t] = 0x7fffffff; }

    for (int m = lane; m < N; m += 32) {
        float dot = 0.f;
#pragma unroll
        for (int c = 0; c < C; ++c) dot += q[c] * Xb[c * N + m];
        float cd = qs + sqb[m] - 2.f * dot;
        int   cm = m;
#pragma unroll
        for (int t = 0; t < TOP; ++t) {
            bool less = (cd < bd[t]) || (cd == bd[t] && cm < bi[t]);
            if (less) { float td = bd[t]; int ti = bi[t]; bd[t] = cd; bi[t] = cm; cd = td; cm = ti; }
        }
    }

    int* orow = idxOut + ((long long)b * N + n) * KNN;
    for (int r = 0; r < TOP; ++r) {
        float vd = bd[0]; int vi = bi[0];
#pragma unroll
        for (int off = 16; off >= 1; off >>= 1) {
            float od = __shfl_xor(vd, off, 32);
            int   oi = __shfl_xor(vi, off, 32);
            if (od < vd || (od == vd && oi < vi)) { vd = od; vi = oi; }
        }
        if (r >= 1 && lane == 0) orow[r - 1] = vi;
        if (bi[0] == vi) {   // owner lane pops its head (m unique across lanes)
#pragma unroll
            for (int t = 0; t < TOP - 1; ++t) { bd[t] = bd[t + 1]; bi[t] = bi[t + 1]; }
            bd[TOP - 1] = 3.4e38f; bi[TOP - 1] = 0x7fffffff;
        }
    }
}

// ---------------------------------------------------------------------------
__global__ void buildxz_kernel(const float* __restrict__ x, const float* __restrict__ z,
                               float* __restrict__ xt, float* __restrict__ xz,
                               int N, int total)
{
    int i = blockIdx.x * blockDim.x + threadIdx.x;
    if (i >= total) return;
    int b = i / N, n = i - b * N;
    const float* xr = x + ((long long)b * N + n) * 3;
    const float* zr = z + ((long long)b * N + n) * 128;
    float* xtb = xt + (long long)b * 3   * N;
    float* xzb = xz + (long long)b * 131 * N;
    for (int c = 0; c < 3; ++c) { float v = xr[c]; xtb[c * N + n] = v; xzb[c * N + n] = v; }
    for (int c = 0; c < 128; ++c) xzb[(3 + c) * N + n] = zr[c];
}

__global__ void sqnorm_kernel(const float* __restrict__ X, float* __restrict__ sq,
                              int C, int N, int total)
{
    int i = blockIdx.x * blockDim.x + threadIdx.x;
    if (i >= total) return;
    int b = i / N, n = i - b * N;
    const float* Xb = X + (long long)b * C * N + n;
    float s = 0.f;
    for (int c = 0; c < C; ++c) { float v = Xb[(long long)c * N]; s += v * v; }
    sq[i] = s;
}

__global__ void softmax_mul_kernel(const float* __restrict__ w, float* __restrict__ h,
                                   int total /*M*N*/, int K)
{
    int i = blockIdx.x * blockDim.x + threadIdx.x;
    if (i >= total) return;
    const float* wr = w + (long long)i * K;
    float*       hr = h + (long long)i * K;
    float mx = wr[0];
    for (int j = 1; j < K; ++j) mx = fmaxf(mx, wr[j]);
    float e[KNN], s = 0.f;
    for (int j = 0; j < K; ++j) { e[j] = expf(wr[j] - mx); s += e[j]; }
    float inv = 1.f / s;
    for (int j = 0; j < K; ++j) hr[j] *= e[j] * inv;
}

// In-place: x <- g * lrelu(x,slope)-instance-normalized + beta, stats over N.
__global__ __launch_bounds__(256) void adain_kernel(float* __restrict__ x,
                                                    const float* __restrict__ s,
                                                    int C, int N, float slope)
{
    const int c = blockIdx.x, b = blockIdx.y;
    float* row      = x + ((long long)b * C + c) * N;
    const float* g  = s + ((long long)b * 2 * C + c) * N;
    const float* be = g + (long long)C * N;
    __shared__ float r1[256], r2[256];
    float sm = 0.f, sqv = 0.f;
    for (int n = threadIdx.x; n < N; n += 256) {
        float v = row[n]; v = (v >= 0.f) ? v : slope * v;
        sm += v; sqv += v * v;
    }
    r1[threadIdx.x] = sm; r2[threadIdx.x] = sqv;
    __syncthreads();
    for (int o = 128; o > 0; o >>= 1) {
        if (threadIdx.x < o) { r1[threadIdx.x] += r1[threadIdx.x + o]; r2[threadIdx.x] += r2[threadIdx.x + o]; }
        __syncthreads();
    }
    float mean = r1[0] / N;
    float var  = r2[0] / N - mean * mean;
    float rs   = rsqrtf(var + 1e-5f);
    for (int n = threadIdx.x; n < N; n += 256) {
        float v = row[n]; v = (v >= 0.f) ? v : slope * v;
        row[n] = g[n] * ((v - mean) * rs) + be[n];
    }
}

__global__ __launch_bounds__(256) void rowmax_kernel(const float* __restrict__ x,
                                                     float* __restrict__ out, int C, int N)
{
    const int c = blockIdx.x, b = blockIdx.y;
    const float* row = x + ((long long)b * C + c) * N;
    __shared__ float r[256];
    float mx = -3.4e38f;
    for (int n = threadIdx.x; n < N; n += 256) mx = fmaxf(mx, row[n]);
    r[threadIdx.x] = mx;
    __syncthreads();
    for (int o = 128; o > 0; o >>= 1) {
        if (threadIdx.x < o) r[threadIdx.x] = fmaxf(r[threadIdx.x], r[threadIdx.x + o]);
        __syncthreads();
    }
    if (threadIdx.x == 0) out[b * C + c] = r[0];
}

__global__ void fc_kernel(const float* __restrict__ W, int Wstride,
                          const float* __restrict__ bias,
                          const float* __restrict__ gamma, const float* __restrict__ beta,
                          float bnr, const float* __restrict__ in, float* __restrict__ out,
                          int Cin, int O, int Btot, float slope)
{
    int i = blockIdx.x * blockDim.x + threadIdx.x;
    if (i >= O * Btot) return;
    int b = i / O, o = i - b * O;
    const float* wr = W  + (long long)o * Wstride;
    const float* ir = in + (long long)b * Cin;
    float y = 0.f;
    for (int c = 0; c < Cin; ++c) y += wr[c] * ir[c];
    if (bias)  y += bias[o];
    if (gamma) y = y * (gamma[o] * bnr) + beta[o];
    out[i] = (y >= 0.f) ? y : slope * y;
}

__global__ void final_kernel(const float* __restrict__ W /*(3,64)*/,
                             const float* __restrict__ b3,
                             const float* __restrict__ t2,
                             float* __restrict__ out, int N, int total)
{
    int i = blockIdx.x * blockDim.x + threadIdx.x;
    if (i >= total) return;
    int b = i / N, n = i - b * N;
    const float* tb = t2 + (long long)b * 64 * N + n;
    float a0 = b3[0], a1 = b3[1], a2 = b3[2];
    for (int c = 0; c < 64; ++c) {
        float v = tb[(long long)c * N];
        a0 += W[c] * v; a1 += W[64 + c] * v; a2 += W[128 + c] * v;
    }
    float* o = out + (long long)i * 3;
    o[0] = a0; o[1] = a1; o[2] = a2;
}

// ---------------------------------------------------------------------------
extern "C" void kernel_launch(void* const* d_in, const int* in_sizes, int n_in,
                              void* d_out, int out_size, void* d_ws, size_t ws_size,
                              hipStream_t stream)
{
    const float* z_in  = (const float*)d_in[0];
    const float* x_in  = (const float*)d_in[1];
    const float* hW1   = (const float*)d_in[2];
    const float* hb1   = (const float*)d_in[3];
    const float* hW2   = (const float*)d_in[4];
    const float* hb2   = (const float*)d_in[5];
    const float* e1_wc1  = (const float*)d_in[6];
    const float* e1_wcb1 = (const float*)d_in[7];
    const float* e1_bn1g = (const float*)d_in[8];
    const float* e1_bn1b = (const float*)d_in[9];
    const float* e1_wc2  = (const float*)d_in[10];
    const float* e1_wcb2 = (const float*)d_in[11];
    const float* e1_bn2g = (const float*)d_in[12];
    const float* e1_bn2b = (const float*)d_in[13];
    const float* e1_xc   = (const float*)d_in[14];
    const float* e1_xcb  = (const float*)d_in[15];
    const float* e1_xbng = (const float*)d_in[16];
    const float* e1_xbnb = (const float*)d_in[17];
    const float* e1_oW   = (const float*)d_in[18];
    const float* e1_ob   = (const float*)d_in[19];
    const float* a1_W    = (const float*)d_in[20];
    const float* a1_b    = (const float*)d_in[21];
    const float* e2_wc1  = (const float*)d_in[22];
    const float* e2_wcb1 = (const float*)d_in[23];
    const float* e2_bn1g = (const float*)d_in[24];
    const float* e2_bn1b = (const float*)d_in[25];
    const float* e2_wc2  = (const float*)d_in[26];
    const float* e2_wcb2 = (const float*)d_in[27];
    const float* e2_bn2g = (const float*)d_in[28];
    const float* e2_bn2b = (const float*)d_in[29];
    const float* e2_xc   = (const float*)d_in[30];
    const float* e2_xcb  = (const float*)d_in[31];
    const float* e2_xbng = (const float*)d_in[32];
    const float* e2_xbnb = (const float*)d_in[33];
    const float* e2_oW   = (const float*)d_in[34];
    const float* e2_ob   = (const float*)d_in[35];
    const float* a2_W    = (const float*)d_in[36];
    const float* a2_b    = (const float*)d_in[37];
    const float* gW1     = (const float*)d_in[38];
    const float* gb1     = (const float*)d_in[39];
    const float* gbn1g   = (const float*)d_in[40];
    const float* gbn1b   = (const float*)d_in[41];
    const float* gW2     = (const float*)d_in[42];
    const float* gb2     = (const float*)d_in[43];
    const float* gbn2g   = (const float*)d_in[44];
    const float* gbn2b   = (const float*)d_in[45];
    const float* tW1     = (const float*)d_in[46];
    const float* tb1     = (const float*)d_in[47];
    const float* tW2     = (const float*)d_in[48];
    const float* tb2     = (const float*)d_in[49];
    const float* tW3     = (const float*)d_in[50];
    const float* tb3     = (const float*)d_in[51];

    const int N = NPTS, B = NBATCH, K = KNN;
    const int NcE = N * K;                 // 20480 edge columns per batch
    const float bnr = 1.0f / sqrtf(1.0f + 1e-5f);

    // ---- workspace layout (floats) ----
    float* ws = (float*)d_ws;
    size_t off = 0;
    auto alloc = [&](size_t nElem) { float* p = ws + off; off += nElem; return p; };
    float* xt    = alloc((size_t)B * 3   * N);
    float* xz    = alloc((size_t)B * 131 * N);
    float* sta   = alloc((size_t)B * 128 * N);
    float* st    = alloc((size_t)B * 128 * N);
    float* sqn   = alloc((size_t)B * N);
    float* x1    = alloc((size_t)B * 64  * N);
    float* x2    = alloc((size_t)B * 128 * N);
    float* sbuf  = alloc((size_t)B * 256 * N);
    float* w1buf = alloc((size_t)64  * NcE);        // per-batch reuse
    float* w2buf = alloc((size_t)128 * NcE);
    float* hbuf  = alloc((size_t)128 * NcE);
    float* fg    = alloc((size_t)B * 128);
    float* fg1   = alloc((size_t)B * 128);
    float* fg2   = alloc((size_t)B * 512);
    float* pb    = alloc((size_t)B * 256);
    float* t1    = alloc((size_t)B * 256 * N);
    float* t2    = alloc((size_t)B * 64  * N);
    int*   idx   = (int*)alloc((size_t)B * N * K);
    (void)ws_size; (void)in_sizes; (void)n_in; (void)out_size;

    auto gemm = [&](int mode, const float* A, int Astride, int Ktot, int M,
                    const float* Bb, long long bBatch, int sK1, int sN, int K2,
                    const float* X, int Cedge, int co, const int* idxp,
                    float* out, long long oBatch, int Nc, int nb,
                    const float* bias, int biasBatch,
                    const float* gamma, const float* beta, float slope) {
        dim3 grid(Nc / 512, M / 16, nb);   // Nc % 512 == 0 and M % 16 == 0 by construction
        if (mode == 0)
            gemm_wmma_kernel<0><<<grid, 256, 0, stream>>>(A, Astride, Ktot, M, Bb, bBatch, sK1, sN, K2,
                X, Cedge, co, N, K, idxp, out, oBatch, Nc, bias, biasBatch, gamma, beta, bnr, slope);
        else if (mode == 1)
            gemm_wmma_kernel<1><<<grid, 256, 0, stream>>>(A, Astride, Ktot, M, Bb, bBatch, sK1, sN, K2,
                X, Cedge, co, N, K, idxp, out, oBatch, Nc, bias, biasBatch, gamma, beta, bnr, slope);
        else
            gemm_wmma_kernel<2><<<grid, 256, 0, stream>>>(A, Astride, Ktot, M, Bb, bBatch, sK1, sN, K2,
                X, Cedge, co, N, K, idxp, out, oBatch, Nc, bias, biasBatch, gamma, beta, bnr, slope);
    };

    const int bnTotal = B * N;
    dim3 blk(256);

    // transpose / concat inputs
    buildxz_kernel<<<(bnTotal + 255) / 256, blk, 0, stream>>>(x_in, z_in, xt, xz, N, bnTotal);

    // style MLP: st = lrelu(hW2 @ lrelu(hW1 @ [x;z] + hb1) + hb2)
    gemm(0, hW1, 131, 131, 128, xz, (long long)131 * N, N, 1, 1, nullptr, 0, 0, nullptr,
         sta, (long long)128 * N, N, B, hb1, 0, nullptr, nullptr, NEG_A);
    gemm(0, hW2, 128, 128, 128, sta, (long long)128 * N, N, 1, 1, nullptr, 0, 0, nullptr,
         st, (long long)128 * N, N, B, hb2, 0, nullptr, nullptr, NEG_A);

    // KNN on raw coordinates (C=3)
    sqnorm_kernel<<<(bnTotal + 255) / 256, blk, 0, stream>>>(xt, sqn, 3, N, bnTotal);
    knn_kernel<3><<<dim3(N, B), 32, 0, stream>>>(xt, sqn, idx, N);

    // edge block 1 (C=3), per-batch scratch reuse
    for (int b = 0; b < B; ++b) {
        const float* Xb = xt + (long long)b * 3 * N;
        const int*   ib = idx + (long long)b * N * K;
        gemm(2, e1_wc1, 3, 3, 32, nullptr, 0, 0, 0, 1, Xb, 3, 3, ib,
             w1buf, 0, NcE, 1, e1_wcb1, 0, e1_bn1g, e1_bn1b, NEG_A);
        gemm(0, e1_wc2, 32, 32, 64, w1buf, 0, NcE, 1, 1, nullptr, 0, 0, nullptr,
             w2buf, 0, NcE, 1, e1_wcb2, 0, e1_bn2g, e1_bn2b, NEG_A);
        gemm(2, e1_xc, 6, 6, 64, nullptr, 0, 0, 0, 1, Xb, 3, 0, ib,
             hbuf, 0, NcE, 1, e1_xcb, 0, e1_xbng, e1_xbnb, NEG_A);
        softmax_mul_kernel<<<(64 * N + 255) / 256, blk, 0, stream>>>(w2buf, hbuf, 64 * N, K);
        gemm(1, e1_oW, 640, 640, 64, hbuf, 0, NcE, K, K, nullptr, 0, 0, nullptr,
             x1 + (long long)b * 64 * N, 0, N, 1, e1_ob, 0, nullptr, nullptr, 1.0f);
    }

    // AdaIN 1: s1 = a1_W @ st + a1_b ; x1 <- adain(lrelu(x1,0.2))
    gemm(0, a1_W, 128, 128, 128, st, (long long)128 * N, N, 1, 1, nullptr, 0, 0, nullptr,
         sbuf, (long long)128 * N, N, B, a1_b, 0, nullptr, nullptr, 1.0f);
    adain_kernel<<<dim3(64, B), blk, 0, stream>>>(x1, sbuf, 64, N, NEG_B);

    // KNN on 64-d features
    sqnorm_kernel<<<(bnTotal + 255) / 256, blk, 0, stream>>>(x1, sqn, 64, N, bnTotal);
    knn_kernel<64><<<dim3(N, B), 32, 0, stream>>>(x1, sqn, idx, N);

    // edge block 2 (C=64)
    for (int b = 0; b < B; ++b) {
        const float* Xb = x1 + (long long)b * 64 * N;
        const int*   ib = idx + (long long)b * N * K;
        gemm(2, e2_wc1, 64, 64, 64, nullptr, 0, 0, 0, 1, Xb, 64, 64, ib,
             w1buf, 0, NcE, 1, e2_wcb1, 0, e2_bn1g, e2_bn1b, NEG_A);
        gemm(0, e2_wc2, 64, 64, 128, w1buf, 0, NcE, 1, 1, nullptr, 0, 0, nullptr,
             w2buf, 0, NcE, 1, e2_wcb2, 0, e2_bn2g, e2_bn2b, NEG_A);
        gemm(2, e2_xc, 128, 128, 128, nullptr, 0, 0, 0, 1, Xb, 64, 0, ib,
             hbuf, 0, NcE, 1, e2_xcb, 0, e2_xbng, e2_xbnb, NEG_A);
        softmax_mul_kernel<<<(128 * N + 255) / 256, blk, 0, stream>>>(w2buf, hbuf, 128 * N, K);
        gemm(1, e2_oW, 1280, 1280, 128, hbuf, 0, NcE, K, K, nullptr, 0, 0, nullptr,
             x2 + (long long)b * 128 * N, 0, N, 1, e2_ob, 0, nullptr, nullptr, 1.0f);
    }

    // AdaIN 2
    gemm(0, a2_W, 128, 128, 256, st, (long long)128 * N, N, 1, 1, nullptr, 0, 0, nullptr,
         sbuf, (long long)256 * N, N, B, a2_b, 0, nullptr, nullptr, 1.0f);
    adain_kernel<<<dim3(128, B), blk, 0, stream>>>(x2, sbuf, 128, N, NEG_B);

    // global feature: max over N, two FC+BN+lrelu layers
    rowmax_kernel<<<dim3(128, B), blk, 0, stream>>>(x2, fg, 128, N);
    fc_kernel<<<(B * 128 + 255) / 256, blk, 0, stream>>>(gW1, 128, gb1, gbn1g, gbn1b, bnr, fg,  fg1, 128, 128, B, NEG_A);
    fc_kernel<<<(B * 512 + 255) / 256, blk, 0, stream>>>(gW2, 128, gb2, gbn2g, gbn2b, bnr, fg1, fg2, 128, 512, B, NEG_A);
    // per-batch bias for tail conv1: pb[b,o] = tb1[o] + tW1[o,:512] . fg2[b]
    fc_kernel<<<(B * 256 + 255) / 256, blk, 0, stream>>>(tW1, 640, tb1, nullptr, nullptr, bnr, fg2, pb, 512, 256, B, 1.0f);

    // tail: t1 = lrelu(tW1[:,512:] @ x2 + pb) ; t2 = lrelu(tW2 @ t1 + tb2) ; out = tW3 @ t2 + tb3
    gemm(0, tW1 + 512, 640, 128, 256, x2, (long long)128 * N, N, 1, 1, nullptr, 0, 0, nullptr,
         t1, (long long)256 * N, N, B, pb, 256, nullptr, nullptr, NEG_A);
    gemm(0, tW2, 256, 256, 64, t1, (long long)256 * N, N, 1, 1, nullptr, 0, 0, nullptr,
         t2, (long long)64 * N, N, B, tb2, 0, nullptr, nullptr, NEG_A);
    final_kernel<<<(bnTotal + 255) / 256, blk, 0, stream>>>(tW3, tb3, t2, (float*)d_out, N, bnTotal);
}